// MemoryAccess_29609504539569
// MI455X (gfx1250) — compile-verified
//
#include <hip/hip_runtime.h>

// ---------------- problem constants ----------------
#define N 512
#define WORD 64
#define R 4
#define CDIM 512
#define B 16
#define T 32
#define IFSZ 471
#define IFP 480          // padded interface width (30 tiles of 16)
#define EPSV 1e-8f

typedef __attribute__((ext_vector_type(2))) float v2f;
typedef __attribute__((ext_vector_type(8))) float v8f;
typedef unsigned int u32x4 __attribute__((ext_vector_type(4)));
typedef int i32x8 __attribute__((ext_vector_type(8)));
typedef int i32x4 __attribute__((ext_vector_type(4)));

// fp32 WMMA: D(16x16) = A(16x4) * B(4x16) + C   (V_WMMA_F32_16X16X4_F32)
__device__ __forceinline__ v8f wmma_f32_4(v2f a, v2f b, v8f c) {
  return __builtin_amdgcn_wmma_f32_16x16x4_f32(false, a, false, b, (short)0, c,
                                               false, false);
}

// ---- Tensor Data Mover: async 2D f32 tile load (global -> LDS) ----
// Builds D# group0/group1 per CDNA5 ISA ch.8 and issues TENSOR_LOAD_TO_LDS.
// This toolchain exposes the 6-arg builtin: (g0, g1, g2, g3, g4, cpol).
// pad_icode: pads after 2^(pad_icode+1) dwords; pad_acode: (pad_acode+1) dwords.
__device__ __forceinline__ void tdm_load_2d(unsigned lds_off, const void* gptr,
                                            unsigned tensor_d0, unsigned tensor_d1,
                                            unsigned tile_d0, unsigned tile_d1,
                                            unsigned d0_stride, bool pad,
                                            unsigned pad_icode, unsigned pad_acode) {
  unsigned long long ga = (unsigned long long)gptr;
  u32x4 g0;
  g0.x = 1u;                                               // count=1 (valid D#)
  g0.y = lds_off;                                          // LDS byte address
  g0.z = (unsigned)(ga & 0xffffffffu);                     // global_addr[31:0]
  g0.w = (unsigned)((ga >> 32) & 0x1ffffffu) | (2u << 30); // addr[56:32] | type=2
  i32x8 g1;
  g1[0] = (int)((2u << 16) |                               // data_size = 4 bytes
                (pad ? (1u << 20) : 0u) |                  // pad_enable
                (pad_icode << 22) | (pad_acode << 25));
  g1[1] = (int)((tensor_d0 & 0xffffu) << 16);              // tensor_dim0[15:0]
  g1[2] = (int)(((tensor_d0 >> 16) & 0xffffu) | ((tensor_d1 & 0xffffu) << 16));
  g1[3] = (int)(((tensor_d1 >> 16) & 0xffffu) | ((tile_d0 & 0xffffu) << 16));
  g1[4] = (int)(tile_d1 & 0xffffu);                        // tile_dim1 (tile_dim2=0)
  g1[5] = (int)d0_stride;                                  // tensor_dim0_stride[31:0]
  g1[6] = 0;
  g1[7] = 0;
  i32x4 z4 = {0, 0, 0, 0};                                 // groups 2/3: 2D tensor
  i32x8 z8 = {0, 0, 0, 0, 0, 0, 0, 0};                     // extra group (unused)
  __builtin_amdgcn_tensor_load_to_lds(g0, g1, z4, z4, z8, 0);
}

__device__ __forceinline__ unsigned lds_offset_of(const void* p) {
  return (unsigned)(unsigned long long)(uintptr_t)p;       // low 32b = LDS offset
}

// ---------------- init ----------------
__global__ void k_init(float* mem, float* link, float* rw, float* prec,
                       float* usage) {
  size_t gid = (size_t)blockIdx.x * blockDim.x + threadIdx.x;
  if (gid < (size_t)B * N * N) link[gid] = 0.f;
  if (gid < (size_t)B * N * WORD) mem[gid] = 0.f;
  if (gid < (size_t)B * R * N) rw[gid] = 1.0f / (float)N;
  if (gid < (size_t)B * N) { prec[gid] = 0.f; usage[gid] = 0.f; }
}

// pad W_if (CDIM x IFSZ) -> Wp (CDIM x IFP), bif -> bp (IFP)
__global__ void k_pad(const float* __restrict__ Wif, const float* __restrict__ bif,
                      float* __restrict__ Wp, float* __restrict__ bp) {
  int gid = blockIdx.x * blockDim.x + threadIdx.x;
  if (gid < IFP) bp[gid] = (gid < IFSZ) ? bif[gid] : 0.f;
  if (gid < CDIM * IFP) {
    int k = gid / IFP, c = gid % IFP;
    Wp[gid] = (c < IFSZ) ? Wif[(size_t)k * IFSZ + c] : 0.f;
  }
}

// ---------------- K1: interface GEMM via f32 WMMA (branch-free) ----------
__global__ void k_iface(const float* __restrict__ ctrl, const float* __restrict__ Wp,
                        const float* __restrict__ bp, float* __restrict__ iface,
                        int t) {
  int tile = blockIdx.x;          // 0..29
  int lane = threadIdx.x;         // wave32
  int m = lane & 15;
  int h = lane >> 4;
  int col = tile * 16 + m;
  const float* arow = ctrl + (size_t)m * T * CDIM + (size_t)t * CDIM;
  v8f acc = {};
  for (int k0 = 0; k0 < CDIM; k0 += 4) {
    int ka = k0 + 2 * h;
    v2f a, b;
    a.x = arow[ka];
    a.y = arow[ka + 1];
    b.x = Wp[(size_t)ka * IFP + col];
    b.y = Wp[(size_t)(ka + 1) * IFP + col];
    acc = wmma_f32_4(a, b, acc);
  }
  float bias = bp[col];
#pragma unroll
  for (int v = 0; v < 8; ++v) {
    int row = v + 8 * h;          // batch index
    iface[(size_t)row * IFP + col] = acc[v] + bias;
  }
}

// ---- shared helper: cosine-similarity softmax over N slots -> simsh ----
__device__ void content_softmax(const float* __restrict__ key, float beta,
                                const float* __restrict__ M, float* keysh,
                                float* red, float* simsh) {
  if (threadIdx.x < WORD) keysh[threadIdx.x] = key[threadIdx.x];
  __syncthreads();
  float kk = 0.f;
#pragma unroll
  for (int w = 0; w < WORD; ++w) kk += keysh[w] * keysh[w];
  float kinv = rsqrtf(kk + EPSV);
  float localmax = -1e30f;
  for (int n = threadIdx.x; n < N; n += blockDim.x) {
    const float* row = M + (size_t)n * WORD;
    if (n + (int)blockDim.x < N)
      __builtin_prefetch(row + (size_t)blockDim.x * WORD, 0, 1);
    float mm = 0.f, dot = 0.f;
#pragma unroll 8
    for (int w = 0; w < WORD; ++w) { float mv = row[w]; mm += mv * mv; dot += keysh[w] * mv; }
    float s = dot * kinv * rsqrtf(mm + EPSV) * beta;
    simsh[n] = s;
    localmax = fmaxf(localmax, s);
  }
  red[threadIdx.x] = localmax; __syncthreads();
  for (int s = 128; s > 0; s >>= 1) {
    if (threadIdx.x < s) red[threadIdx.x] = fmaxf(red[threadIdx.x], red[threadIdx.x + s]);
    __syncthreads();
  }
  float mx = red[0]; __syncthreads();
  float lsum = 0.f;
  for (int n = threadIdx.x; n < N; n += blockDim.x) {
    float e = expf(simsh[n] - mx); simsh[n] = e; lsum += e;
  }
  red[threadIdx.x] = lsum; __syncthreads();
  for (int s = 128; s > 0; s >>= 1) {
    if (threadIdx.x < s) red[threadIdx.x] += red[threadIdx.x + s];
    __syncthreads();
  }
  float inv = 1.f / red[0]; __syncthreads();
  for (int n = threadIdx.x; n < N; n += blockDim.x) simsh[n] *= inv;
  __syncthreads();
}

// ---------------- K2: fused per-batch step -------------------------------
// parse activations -> write-content softmax -> allocation (bitonic argsort
// + product scan) -> write weights / usage update / sum(ww)
__global__ void k_batch(float* __restrict__ iface, const float* __restrict__ mem,
                        const float* __restrict__ rw, float* __restrict__ ww,
                        float* __restrict__ usage, float* __restrict__ Ssum) {
  int b = blockIdx.x;
  float* p = iface + (size_t)b * IFP;
  __shared__ float keysh[WORD];
  __shared__ float red[256];
  __shared__ float simsh[N];      // write-content weights
  __shared__ float val[N];
  __shared__ unsigned short sidx[N];
  __shared__ float scan[N];
  __shared__ float allocsh[N];

  // 1) parse
  for (int i = threadIdx.x; i < IFSZ; i += blockDim.x) {
    float x = p[i];
    if ((i >= 256 && i < 260) || i == 324) {
      p[i] = (x > 20.f) ? x : log1pf(expf(x));          // softplus
    } else if ((i >= 325 && i < 389) || (i >= 453 && i < 459)) {
      p[i] = 1.f / (1.f + expf(-x));                     // sigmoid
    }
  }
  __syncthreads();
  if (threadIdx.x < R) {                                 // read-mode softmax
    int base = 459 + threadIdx.x * 3;
    float a = p[base], b2 = p[base + 1], c2 = p[base + 2];
    float mx = fmaxf(a, fmaxf(b2, c2));
    float ea = expf(a - mx), eb = expf(b2 - mx), ec = expf(c2 - mx);
    float s = ea + eb + ec;
    p[base] = ea / s; p[base + 1] = eb / s; p[base + 2] = ec / s;
  }
  __syncthreads();

  // 2) write-content weights (old memory)
  content_softmax(p + 260, p[324], mem + (size_t)b * N * WORD, keysh, red, simsh);

  // 3) allocation from OLD usage
  for (int i = threadIdx.x; i < N; i += blockDim.x) {
    val[i] = usage[(size_t)b * N + i];
    sidx[i] = (unsigned short)i;
  }
  __syncthreads();
  for (int k = 2; k <= N; k <<= 1) {
    for (int j = k >> 1; j > 0; j >>= 1) {
      for (int e = threadIdx.x; e < N; e += blockDim.x) {
        int l = e ^ j;
        if (l > e) {
          bool up = ((e & k) == 0);
          float v1 = val[e], v2 = val[l];
          unsigned short i1 = sidx[e], i2 = sidx[l];
          bool sw = up ? (v1 > v2 || (v1 == v2 && i1 > i2))
                       : (v1 < v2 || (v1 == v2 && i1 < i2));
          if (sw) { val[e] = v2; val[l] = v1; sidx[e] = i2; sidx[l] = i1; }
        }
      }
      __syncthreads();
    }
  }
  int e0 = threadIdx.x, e1 = threadIdx.x + 256;
  scan[e0] = val[e0]; scan[e1] = val[e1];
  __syncthreads();
  for (int off = 1; off < N; off <<= 1) {
    float t0 = (e0 >= off) ? scan[e0 - off] * scan[e0] : scan[e0];
    float t1 = (e1 >= off) ? scan[e1 - off] * scan[e1] : scan[e1];
    __syncthreads();
    scan[e0] = t0; scan[e1] = t1;
    __syncthreads();
  }
  allocsh[sidx[e0]] = (1.f - val[e0]) * ((e0 == 0) ? 1.f : scan[e0 - 1]);
  allocsh[sidx[e1]] = (1.f - val[e1]) * scan[e1 - 1];
  __syncthreads();

  // 4) write weights, usage update, sum(ww)
  float ag = p[457], wg = p[458];
  float fg0 = p[453], fg1 = p[454], fg2 = p[455], fg3 = p[456];
  float lsum = 0.f;
  for (int n = threadIdx.x; n < N; n += blockDim.x) {
    float w = wg * (ag * allocsh[n] + (1.f - ag) * simsh[n]);
    ww[(size_t)b * N + n] = w;
    lsum += w;
    float u0 = usage[(size_t)b * N + n];
    float u = u0 + w - u0 * w;
    float psi = (1.f - fg0 * rw[((size_t)b * R + 0) * N + n]) *
                (1.f - fg1 * rw[((size_t)b * R + 1) * N + n]) *
                (1.f - fg2 * rw[((size_t)b * R + 2) * N + n]) *
                (1.f - fg3 * rw[((size_t)b * R + 3) * N + n]);
    usage[(size_t)b * N + n] = u * psi;
  }
  red[threadIdx.x] = lsum; __syncthreads();
  for (int s = 128; s > 0; s >>= 1) {
    if (threadIdx.x < s) red[threadIdx.x] += red[threadIdx.x + s];
    __syncthreads();
  }
  if (threadIdx.x == 0) Ssum[b] = red[0];
}

// ---------------- K3: memory erase+write ----------------
__global__ void k_mem(const float* __restrict__ iface, const float* __restrict__ ww,
                      float* __restrict__ mem) {
  size_t gid = (size_t)blockIdx.x * blockDim.x + threadIdx.x;  // B*N*WORD
  int w = gid % WORD;
  int n = (gid / WORD) % N;
  int b = gid / ((size_t)WORD * N);
  const float* p = iface + (size_t)b * IFP;
  float e = p[325 + w], wv = p[389 + w];
  float wn = ww[(size_t)b * N + n];
  mem[gid] = mem[gid] * (1.f - wn * e) + wn * wv;
}

// ---------------- K4: link update fused with fwd matvec (TDM + WMMA) -----
// TDM streams the 16x512 row tile into LDS (padded 2 dwords / 256 dwords so
// LDS row stride = 516 dwords -> conflict-free B-operand reads).
__global__ void k_link_fwd(float* __restrict__ link, const float* __restrict__ ww,
                           const float* __restrict__ prec,
                           const float* __restrict__ rw, float* __restrict__ fwd) {
  __shared__ float tile[16 * 516 + 8];   // 8192 dwords + 64 pad dwords
  int blk = blockIdx.x;
  int b = blk / (N / 16);
  int it = blk % (N / 16);
  int lane = threadIdx.x;
  int m = lane & 15, h = lane >> 4;
  float* L = link + (size_t)b * N * N;
  const float* rowbase = L + (size_t)it * 16 * N;
  tdm_load_2d(lds_offset_of(&tile[0]), rowbase, N, N,
              /*tile_d0=*/N, /*tile_d1=*/16, /*stride=*/N,
              /*pad=*/true, /*interval 2^8=256dw*/ 7u, /*amount 2dw*/ 1u);

  int i = it * 16 + m;
  float wsi = ww[(size_t)b * N + i];
  const float* rwb = rw + (size_t)b * R * N;
  const float* wwb = ww + (size_t)b * N;
  const float* pb = prec + (size_t)b * N;
  __builtin_amdgcn_s_wait_tensorcnt(0);

  v8f acc = {};
  for (int j0 = 0; j0 < N; j0 += 4) {
    int ja = j0 + 2 * h;
    int pidx = m * 516 + ja + 2 * (ja >> 8);   // padded LDS dword address
    float l0 = tile[pidx];
    float l1 = tile[pidx + 1];
    float nl0 = (1.f - wsi - wwb[ja]) * l0 + wsi * pb[ja];
    float nl1 = (1.f - wsi - wwb[ja + 1]) * l1 + wsi * pb[ja + 1];
    if (i == ja) nl0 = 0.f;
    if (i == ja + 1) nl1 = 0.f;
    L[(size_t)i * N + ja] = nl0;
    L[(size_t)i * N + ja + 1] = nl1;
    v2f a, bb;
    a.x = (m < R) ? rwb[(size_t)m * N + ja] : 0.f;
    a.y = (m < R) ? rwb[(size_t)m * N + ja + 1] : 0.f;
    bb.x = nl0; bb.y = nl1;
    acc = wmma_f32_4(a, bb, acc);
  }
  if (h == 0) {
#pragma unroll
    for (int v = 0; v < R; ++v)
      fwd[((size_t)b * R + v) * N + it * 16 + m] = acc[v];
  }
}

// ---------------- K5: bwd matvec over NEW link (TDM + WMMA) + prec -------
// TDM gathers the 512x16 COLUMN tile (one strided DMA instead of 512
// stride-2KB loads), then bwd[r][j] = sum_i rw[r][i]*nl[i][j] via WMMA.
__global__ void k_bwd(const float* __restrict__ link, const float* __restrict__ rw,
                      float* __restrict__ bwd, const float* __restrict__ ww,
                      const float* __restrict__ Ssum, float* __restrict__ prec) {
  __shared__ float tile[N * 16];         // 8192 dwords, no pad needed
  int blk = blockIdx.x;
  int b = blk / (N / 16);
  int jt = blk % (N / 16);
  int lane = threadIdx.x;
  int m = lane & 15, h = lane >> 4;
  const float* L = link + (size_t)b * N * N;
  tdm_load_2d(lds_offset_of(&tile[0]), L + (size_t)jt * 16, N, N,
              /*tile_d0=*/16, /*tile_d1=*/N, /*stride=*/N,
              /*pad=*/false, 0u, 0u);

  int j = jt * 16 + m;
  const float* rwb = rw + (size_t)b * R * N;
  __builtin_amdgcn_s_wait_tensorcnt(0);

  v8f acc = {};
  for (int i0 = 0; i0 < N; i0 += 4) {
    int ia = i0 + 2 * h;
    v2f a, bb;
    a.x = (m < R) ? rwb[(size_t)m * N + ia] : 0.f;
    a.y = (m < R) ? rwb[(size_t)m * N + ia + 1] : 0.f;
    bb.x = tile[ia * 16 + m];
    bb.y = tile[(ia + 1) * 16 + m];
    acc = wmma_f32_4(a, bb, acc);
  }
  if (h == 0) {
#pragma unroll
    for (int v = 0; v < R; ++v)
      bwd[((size_t)b * R + v) * N + jt * 16 + m] = acc[v];
    // fused precedence update (link pass already consumed old prec)
    float S = Ssum[b];
    prec[(size_t)b * N + j] = (1.f - S) * prec[(size_t)b * N + j] + ww[(size_t)b * N + j];
  }
}

// ---------------- K6: fused read path ------------------------------------
// read-content softmax (new memory) + mode mix + read words
__global__ void k_read(const float* __restrict__ iface, const float* __restrict__ fwd,
                       const float* __restrict__ bwd, const float* __restrict__ mem,
                       float* __restrict__ rw, float* __restrict__ out, int t) {
  int bh = blockIdx.x, b = bh / R, r = bh % R;
  const float* p = iface + (size_t)b * IFP;
  __shared__ float keysh[WORD];
  __shared__ float red[256];
  __shared__ float simsh[N];      // read-content weights
  __shared__ float rwsh[N];
  const float* M = mem + (size_t)b * N * WORD;

  content_softmax(p + r * WORD, p[256 + r], M, keysh, red, simsh);

  float m0 = p[459 + r * 3], m1 = p[459 + r * 3 + 1], m2 = p[459 + r * 3 + 2];
  for (int n = threadIdx.x; n < N; n += blockDim.x) {
    float v = m0 * bwd[(size_t)bh * N + n] + m1 * simsh[n] + m2 * fwd[(size_t)bh * N + n];
    rw[(size_t)bh * N + n] = v;
    rwsh[n] = v;
  }
  __syncthreads();
  for (int w = threadIdx.x; w < WORD; w += blockDim.x) {
    float s = 0.f;
    for (int n = 0; n < N; ++n) s += rwsh[n] * M[(size_t)n * WORD + w];
    out[(((size_t)b * T + t) * R + r) * WORD + w] = s;
  }
}

// ---------------- host-side launch ----------------
extern "C" void kernel_launch(void* const* d_in, const int* in_sizes, int n_in,
                              void* d_out, int out_size, void* d_ws, size_t ws_size,
                              hipStream_t stream) {
  const float* ctrl = (const float*)d_in[0];
  const float* Wif = (const float*)d_in[1];
  const float* bif = (const float*)d_in[2];
  float* out = (float*)d_out;

  float* ws = (float*)d_ws;
  float* mem = ws;    ws += (size_t)B * N * WORD;
  float* link = ws;   ws += (size_t)B * N * N;
  float* rw = ws;     ws += (size_t)B * R * N;
  float* prec = ws;   ws += (size_t)B * N;
  float* usage = ws;  ws += (size_t)B * N;
  float* iface = ws;  ws += (size_t)B * IFP;
  float* ww = ws;     ws += (size_t)B * N;
  float* fwd = ws;    ws += (size_t)B * R * N;
  float* bwd = ws;    ws += (size_t)B * R * N;
  float* Ssum = ws;   ws += B;
  float* Wp = ws;     ws += (size_t)CDIM * IFP;
  float* bp = ws;     ws += IFP;

  k_init<<<(B * N * N + 255) / 256, 256, 0, stream>>>(mem, link, rw, prec, usage);
  k_pad<<<(CDIM * IFP + 255) / 256, 256, 0, stream>>>(Wif, bif, Wp, bp);

  for (int t = 0; t < T; ++t) {
    k_iface<<<IFP / 16, 32, 0, stream>>>(ctrl, Wp, bp, iface, t);
    k_batch<<<B, 256, 0, stream>>>(iface, mem, rw, ww, usage, Ssum);
    k_mem<<<(B * N * WORD + 255) / 256, 256, 0, stream>>>(iface, ww, mem);
    k_link_fwd<<<B * (N / 16), 32, 0, stream>>>(link, ww, prec, rw, fwd);
    k_bwd<<<B * (N / 16), 32, 0, stream>>>(link, rw, bwd, ww, Ssum, prec);
    k_read<<<B * R, 256, 0, stream>>>(iface, fwd, bwd, mem, rw, out, t);
  }
}